// Conv_layer_attention_80496277061961
// MI455X (gfx1250) — compile-verified
//
#include <hip/hip_runtime.h>

// ---------------------------------------------------------------------------
// Conv_layer_attention for MI455X (gfx1250, wave32, WMMA)
//
// Shapes: bs=4, v=4096, n=50, in_c=out_c=64, sup=4  ->  NCOLS=(4+1)*64=320
// out = feature_center + sum_{j,s} w_j * relu(ndn@sdn)[j,s*64+c] * fs[idx_j, 64+s*64+c]
// (the l1/l2 MLP in the reference is dead code and is skipped)
// ---------------------------------------------------------------------------

typedef __attribute__((ext_vector_type(2))) float v2f;
typedef __attribute__((ext_vector_type(8))) float v8f;

#define BS     4
#define VPB    4096
#define NNEI   50
#define IN_C   64
#define OUT_C  64
#define SUP    4
#define NCOLS  320      // (SUP+1)*OUT_C
#define EPSF   1e-12f

// ---------------------------------------------------------------------------
// Kernel 1: feature_out[16384 x 320] = feature_map[16384 x 64] @ W[64 x 320] + bias
// One wave per 16x16 output tile, V_WMMA_F32_16X16X4_F32, K stepped by 4.
// f32 A layout (16x4): lane%16 = M row; VGPR0/1 hold K = 2*(lane/16), +1.
// f32 B layout (4x16): lane%16 = N col; VGPR0/1 hold K = 2*(lane/16), +1.
// f32 C/D layout (16x16): VGPR g holds M = g + 8*(lane/16), N = lane%16.
// ---------------------------------------------------------------------------
__global__ __launch_bounds__(256)
void feat_gemm_wmma(const float* __restrict__ A,     // [M x 64]
                    const float* __restrict__ W,     // [64 x 320]
                    const float* __restrict__ bias,  // [320]
                    float* __restrict__ out)         // [M x 320]
{
    const int lane = threadIdx.x & 31;
    const int wave = threadIdx.x >> 5;
    const int tile = blockIdx.x * 8 + wave;          // 1024*20 tiles total
    const int mtile = tile / (NCOLS / 16);
    const int ntile = tile % (NCOLS / 16);

    const int l16 = lane >> 4;                       // 0 or 1
    const int lm  = lane & 15;
    const int row = mtile * 16 + lm;                 // A row for this lane
    const int col = ntile * 16 + lm;                 // B/C col for this lane

    v8f c = {};
    #pragma unroll
    for (int k = 0; k < IN_C; k += 4) {
        v2f a, b;
        a.x = A[row * IN_C + k + 2 * l16];
        a.y = A[row * IN_C + k + 2 * l16 + 1];
        b.x = W[(k + 2 * l16) * NCOLS + col];
        b.y = W[(k + 2 * l16 + 1) * NCOLS + col];
        c = __builtin_amdgcn_wmma_f32_16x16x4_f32(false, a, false, b,
                                                  (short)0, c, false, false);
    }

    const float bb = bias[col];
    #pragma unroll
    for (int g = 0; g < 8; ++g) {
        const int r = mtile * 16 + g + 8 * l16;
        out[r * NCOLS + col] = c[g] + bb;
    }
}

// ---------------------------------------------------------------------------
// Kernel 2: one block (256 threads = 8 waves) per vertex.
//   Phase A: build normalized sdn (3x256 -> 4x256 padded), ndn (50x3 -> 64x4
//            padded), normalized neighbor weights w, indices -> LDS.
//   Phase B (x2 halves of 128 channels): theta = relu(ndn @ sdn) via WMMA
//            f32 16x16x4 (single K step, K=3 zero-padded to 4) -> LDS.
//   Phase C: coalesced gather-multiply-reduce over neighbors/supports.
// ---------------------------------------------------------------------------
__global__ __launch_bounds__(256)
void conv_attn(const int*   __restrict__ nbr_idx,   // [bs*v*50]
               const float* __restrict__ verts,     // [bs*v*3]
               const float* __restrict__ nval,      // [bs*v*50]
               const float* __restrict__ dirs,      // [3*256]
               const float* __restrict__ fo,        // [bs*v x 320] (kernel 1)
               float*       __restrict__ out)       // [bs*v x 64]
{
    __shared__ float s_sdn[4][256];      // 4 KB  (row 3 zero pad for K=4)
    __shared__ float s_nd[64][4];        // 1 KB  (rows 50..63 zero, col 3 zero)
    __shared__ float s_w[64];
    __shared__ int   s_idx[64];
    __shared__ float s_theta[64][128];   // 32 KB (one 128-channel half)
    __shared__ float s_red[4][64];       // 1 KB

    const int tid = threadIdx.x;
    const int gv  = blockIdx.x;          // global vertex row = b*4096 + v
    const int b   = gv / VPB;

    // --- sdn: normalize directions over axis 0 (each 3-vector column) ------
    {
        const float d0 = dirs[0 * 256 + tid];
        const float d1 = dirs[1 * 256 + tid];
        const float d2 = dirs[2 * 256 + tid];
        const float inv = 1.0f / fmaxf(sqrtf(d0*d0 + d1*d1 + d2*d2), EPSF);
        s_sdn[0][tid] = d0 * inv;
        s_sdn[1][tid] = d1 * inv;
        s_sdn[2][tid] = d2 * inv;
        s_sdn[3][tid] = 0.0f;
    }

    // --- ndn, neighbor indices, raw neighbor_value -------------------------
    const float vx = verts[gv * 3 + 0];
    const float vy = verts[gv * 3 + 1];
    const float vz = verts[gv * 3 + 2];
    if (tid < 64) {
        if (tid < NNEI) {
            const int idx = nbr_idx[gv * NNEI + tid];
            s_idx[tid] = idx;
            const int nrow = b * VPB + idx;
            const float dx = verts[nrow * 3 + 0] - vx;
            const float dy = verts[nrow * 3 + 1] - vy;
            const float dz = verts[nrow * 3 + 2] - vz;
            const float inv = 1.0f / fmaxf(sqrtf(dx*dx + dy*dy + dz*dz), EPSF);
            s_nd[tid][0] = dx * inv;
            s_nd[tid][1] = dy * inv;
            s_nd[tid][2] = dz * inv;
            s_nd[tid][3] = 0.0f;
            s_w[tid] = nval[gv * NNEI + tid];      // raw, normalized below
        } else {
            s_nd[tid][0] = s_nd[tid][1] = s_nd[tid][2] = s_nd[tid][3] = 0.0f;
            s_w[tid] = 0.0f;
            s_idx[tid] = 0;
        }
    }
    __syncthreads();

    // --- normalize w over the 50 neighbors ---------------------------------
    if (tid == 0) {
        float ss = 0.0f;
        for (int j = 0; j < NNEI; ++j) ss += s_w[j] * s_w[j];
        s_red[0][0] = 1.0f / fmaxf(sqrtf(ss), EPSF);
    }
    __syncthreads();
    const float invn = s_red[0][0];
    if (tid < NNEI) s_w[tid] *= invn;

    const int lane = tid & 31;
    const int wv   = tid >> 5;
    const int l16  = lane >> 4;
    const int lm   = lane & 15;
    const int cch  = tid & 63;           // output channel 0..63
    const int q    = tid >> 6;           // neighbor stripe 0..3

    float acc = 0.0f;

    #pragma unroll
    for (int h = 0; h < 2; ++h) {        // two 128-channel halves
        __syncthreads();                 // s_theta free to reuse; s_w ready

        // --- theta tiles via WMMA: 4 mtiles x 8 ntiles, 8 waves x 4 tiles --
        for (int t = wv; t < 32; t += 8) {
            const int mt = t >> 3;                       // 0..3  (neighbor rows)
            const int nt = t & 7;                        // 0..7  (16-ch cols)
            const int colg = h * 128 + nt * 16 + lm;     // channel in 0..255
            v2f a, bb;
            a.x  = s_nd[mt * 16 + lm][2 * l16];
            a.y  = s_nd[mt * 16 + lm][2 * l16 + 1];
            bb.x = s_sdn[2 * l16][colg];
            bb.y = s_sdn[2 * l16 + 1][colg];
            v8f cc = {};
            cc = __builtin_amdgcn_wmma_f32_16x16x4_f32(false, a, false, bb,
                                                       (short)0, cc, false, false);
            #pragma unroll
            for (int g = 0; g < 8; ++g) {
                const int r = mt * 16 + g + 8 * l16;
                s_theta[r][nt * 16 + lm] = fmaxf(cc[g], 0.0f);   // relu
            }
        }
        __syncthreads();

        // --- gather-multiply-reduce: 64 channels x 4 neighbor stripes ------
        for (int j = q; j < NNEI; j += 4) {
            const float wj = s_w[j];
            const float* fr = fo + (size_t)(b * VPB + s_idx[j]) * NCOLS + OUT_C;
            #pragma unroll
            for (int s2 = 0; s2 < 2; ++s2) {
                const int sIdx = h * 2 + s2;             // support 0..3
                const float th = s_theta[j][s2 * 64 + cch];
                acc += wj * th * fr[sIdx * 64 + cch];
            }
        }
    }

    // --- reduce the 4 stripes, add feature_center, store -------------------
    __syncthreads();
    s_red[q][cch] = acc;
    __syncthreads();
    if (tid < 64) {
        const float fc = fo[(size_t)gv * NCOLS + cch];
        out[(size_t)gv * OUT_C + cch] =
            fc + s_red[0][cch] + s_red[1][cch] + s_red[2][cch] + s_red[3][cch];
    }
}

// ---------------------------------------------------------------------------
extern "C" void kernel_launch(void* const* d_in, const int* in_sizes, int n_in,
                              void* d_out, int out_size, void* d_ws, size_t ws_size,
                              hipStream_t stream) {
    (void)in_sizes; (void)n_in; (void)out_size; (void)ws_size;

    const int*   nbr_idx = (const int*)  d_in[0];
    const float* verts   = (const float*)d_in[1];
    const float* fmap    = (const float*)d_in[2];
    const float* nval    = (const float*)d_in[3];
    const float* W       = (const float*)d_in[4];
    const float* bias    = (const float*)d_in[5];
    const float* dirs    = (const float*)d_in[6];
    // d_in[7..10] (l1_w, l1_b, l2_w, l2_b) are dead code in the reference.

    float* fo = (float*)d_ws;            // [16384 x 320] f32 = ~21 MB scratch
    const int M = BS * VPB;              // 16384

    // 1024 mtiles * 20 ntiles = 20480 tiles, 8 waves/block -> 2560 blocks
    feat_gemm_wmma<<<(M / 16) * (NCOLS / 16) / 8, 256, 0, stream>>>(fmap, W, bias, fo);

    conv_attn<<<M, 256, 0, stream>>>(nbr_idx, verts, nval, dirs, fo, (float*)d_out);
}